// Output_4887672783368
// MI455X (gfx1250) — compile-verified
//
#include <hip/hip_runtime.h>

// MI455X (gfx1250) implementation of banded-softmax disparity regression.
//   K1: colsum/valid-mask over cost_left2right  (streams 134 MB, WMMA column reduce)
//   K2: disp_ini over cost_right2left           (streams 134 MB)
//   K3: iterative hole-filling propagation      (1 wave / row, shfl neighbors)
// Bandwidth-bound: ~268 MB total -> ~11.5 us floor at 23.3 TB/s.

typedef float v2f __attribute__((ext_vector_type(2)));
typedef float v8f __attribute__((ext_vector_type(8)));

#define WW 256
#define EPS_SUM 1e-8f
#define EPS_DIV 1e-4f

__device__ __forceinline__ float wave_max32(float x) {
#pragma unroll
  for (int o = 16; o > 0; o >>= 1) x = fmaxf(x, __shfl_xor(x, o, 32));
  return x;
}
__device__ __forceinline__ float wave_sum32(float x) {
#pragma unroll
  for (int o = 16; o > 0; o >>= 1) x += __shfl_xor(x, o, 32);
  return x;
}

// ---------------------------------------------------------------------------
// Kernel 1: att_l2r banded softmax + column sums -> valid mask (0/1 float).
// One block per (b,h). 8 waves, each wave owns rows {w, w+8, ...} (32 rows).
// Lane l stages cols {4l..4l+3, 128+4l..128+4l+3} in registers (b128 loads).
// Cross-wave column reduction done with V_WMMA_F32_16X16X4_F32 (A = ones).
// ---------------------------------------------------------------------------
__global__ void __launch_bounds__(256) colsum_vm_kernel(
    const float* __restrict__ cost, const int* __restrict__ mdp,
    float* __restrict__ vmbuf) {
  const int bh   = blockIdx.x;
  const int tid  = threadIdx.x;
  const int wave = tid >> 5;
  const int lane = tid & 31;
  const int maxd = *mdp;

  __shared__ float part[8][WW];  // per-wave partial column sums

  int col[8];
#pragma unroll
  for (int t = 0; t < 4; ++t) { col[t] = 4 * lane + t; col[t + 4] = 128 + 4 * lane + t; }

  float acc[8];
#pragma unroll
  for (int t = 0; t < 8; ++t) acc[t] = 0.0f;

  const float* base = cost + (size_t)bh * WW * WW;

  for (int i = wave; i < WW; i += 8) {
    const float4 c0 = *(const float4*)(base + (size_t)i * WW + 4 * lane);
    const float4 c1 = *(const float4*)(base + (size_t)i * WW + 128 + 4 * lane);
    if (i + 8 < WW)  // stream-ahead hint (global_prefetch_b8)
      __builtin_prefetch(base + (size_t)(i + 8) * WW + 4 * lane, 0, 0);

    float cv[8] = {c0.x, c0.y, c0.z, c0.w, c1.x, c1.y, c1.z, c1.w};
    bool  bd[8];
    float m = 0.0f;  // zero-filled row always contributes 0 to the max
#pragma unroll
    for (int t = 0; t < 8; ++t) {
      const int j = col[t];
      bd[t] = (j >= i) && (maxd <= 0 || j < i + maxd);
      if (bd[t]) m = fmaxf(m, cv[t]);
    }
    m = wave_max32(m);

    float a[8], s = 0.0f;
#pragma unroll
    for (int t = 0; t < 8; ++t) {
      a[t] = bd[t] ? __expf(cv[t] - m) : 0.0f;
      s += a[t];
    }
    s = wave_sum32(s);
    const float inv = 1.0f / (s + EPS_SUM);
#pragma unroll
    for (int t = 0; t < 8; ++t) acc[t] += a[t] * inv;
  }

#pragma unroll
  for (int t = 0; t < 8; ++t) part[wave][col[t]] = acc[t];
  __syncthreads();

  if (tid < 32) {  // wave 0, EXEC all ones (WMMA requirement)
    const int ln = tid & 15;
    const int hi = tid >> 4;  // 0 -> lanes 0-15, 1 -> lanes 16-31
    v2f aone; aone[0] = 1.0f; aone[1] = 1.0f;  // A = ones(16x4): layout-invariant
    for (int cb = 0; cb < WW; cb += 16) {
      v2f b1, b2;  // B = 4x16 slab of per-wave partials; sum over K is what we want
      b1[0] = part[hi ? 2 : 0][cb + ln];
      b1[1] = part[hi ? 3 : 1][cb + ln];
      b2[0] = part[hi ? 6 : 4][cb + ln];
      b2[1] = part[hi ? 7 : 5][cb + ln];
      v8f c = {0.f, 0.f, 0.f, 0.f, 0.f, 0.f, 0.f, 0.f};
      c = __builtin_amdgcn_wmma_f32_16x16x4_f32(false, aone, false, b1, (short)0, c, false, false);
      c = __builtin_amdgcn_wmma_f32_16x16x4_f32(false, aone, false, b2, (short)0, c, false, false);
      const float cs = c[0];  // D row 0: lanes 0-15 hold columns cb..cb+15
      if (hi == 0)
        vmbuf[(size_t)bh * WW + cb + ln] = (cs > 0.1f) ? 1.0f : 0.0f;
    }
  }
}

// ---------------------------------------------------------------------------
// Kernel 2: att_r2l banded softmax row reductions -> disp_ini.
// One wave per output row (b,h,j); 8 waves/block.
// ---------------------------------------------------------------------------
__global__ void __launch_bounds__(256) dispini_kernel(
    const float* __restrict__ cost, const int* __restrict__ mdp,
    float* __restrict__ dini) {
  const int tid  = threadIdx.x;
  const int wave = tid >> 5;
  const int lane = tid & 31;
  const int r    = blockIdx.x * 8 + wave;  // flat (b,h,j)
  const int j    = r & (WW - 1);
  const int maxd = *mdp;

  const float* row = cost + (size_t)r * WW;
  const float4 c0 = *(const float4*)(row + 4 * lane);
  const float4 c1 = *(const float4*)(row + 128 + 4 * lane);

  float cv[8] = {c0.x, c0.y, c0.z, c0.w, c1.x, c1.y, c1.z, c1.w};
  int   col[8];
  bool  bd[8];
  float m = 0.0f;
#pragma unroll
  for (int t = 0; t < 4; ++t) { col[t] = 4 * lane + t; col[t + 4] = 128 + 4 * lane + t; }
#pragma unroll
  for (int t = 0; t < 8; ++t) {
    const int k = col[t];
    bd[t] = (k <= j) && (maxd <= 0 || k > j - maxd);
    if (bd[t]) m = fmaxf(m, cv[t]);
  }
  m = wave_max32(m);

  float s = 0.0f, wt = 0.0f;
#pragma unroll
  for (int t = 0; t < 8; ++t) {
    const float a = bd[t] ? __expf(cv[t] - m) : 0.0f;
    s += a;
    wt += a * (float)col[t];
  }
  s  = wave_sum32(s);
  wt = wave_sum32(wt);
  if (lane == 0) dini[r] = (float)j - wt / (s + EPS_SUM);
}

// ---------------------------------------------------------------------------
// Kernel 3: hole-filling propagation (conv [1,1,0] L->R then [0,1,1] R->L).
// One wave per (b,h) row; lane l holds contiguous cols 8l..8l+7.
// Each step grows the valid region by <=1 px; extra reference iterations are
// exact no-ops, so we exit at the fix-point (ballot), capped at W steps.
// ---------------------------------------------------------------------------
__global__ void __launch_bounds__(256) propagate_kernel(
    const float* __restrict__ dini_g, const float* __restrict__ vm_g,
    float* __restrict__ out, int nrows) {
  const int tid  = threadIdx.x;
  const int wave = tid >> 5;
  const int lane = tid & 31;
  const int row  = blockIdx.x * 8 + wave;
  if (row >= nrows) return;  // wave-uniform

  const float* dp = dini_g + (size_t)row * WW + 8 * lane;
  const float* vp = vm_g   + (size_t)row * WW + 8 * lane;
  const float4 d0 = *(const float4*)(dp);
  const float4 d1 = *(const float4*)(dp + 4);
  const float4 m0 = *(const float4*)(vp);
  const float4 m1 = *(const float4*)(vp + 4);

  float di[8] = {d0.x, d0.y, d0.z, d0.w, d1.x, d1.y, d1.z, d1.w};
  float v0[8] = {m0.x, m0.y, m0.z, m0.w, m1.x, m1.y, m1.z, m1.w};
  float v[8], d[8];
#pragma unroll
  for (int t = 0; t < 8; ++t) { v[t] = v0[t]; d[t] = di[t] * v0[t]; }

  // conv1: out[j] = x[j-1] + x[j]
  for (int it = 0; it < WW; ++it) {
    float lv = __shfl_up(v[7], 1, 32);
    float ld = __shfl_up(d[7], 1, 32);
    if (lane == 0) { lv = 0.0f; ld = 0.0f; }
    float nv[8], nd[8];
    bool  ch = false;
#pragma unroll
    for (int t = 0; t < 8; ++t) {
      const float plv = t ? v[t - 1] : lv;
      const float pld = t ? d[t - 1] : ld;
      const float vm1 = plv + v[t];
      const float grow = (vm1 > 0.0f) ? 1.0f : 0.0f;
      nd[t] = d[t] * v[t] +
              (pld + d[t]) * __builtin_amdgcn_rcpf(vm1 + EPS_DIV) * (grow - v[t]);
      nv[t] = grow;
      ch |= (grow != v[t]);
    }
#pragma unroll
    for (int t = 0; t < 8; ++t) { v[t] = nv[t]; d[t] = nd[t]; }
    if (__ballot(ch) == 0) break;
  }

  // conv2: out[j] = x[j] + x[j+1]
  for (int it = 0; it < WW; ++it) {
    float rv = __shfl_down(v[0], 1, 32);
    float rd = __shfl_down(d[0], 1, 32);
    if (lane == 31) { rv = 0.0f; rd = 0.0f; }
    float nv[8], nd[8];
    bool  ch = false;
#pragma unroll
    for (int t = 0; t < 8; ++t) {
      const float prv = (t < 7) ? v[t + 1] : rv;
      const float prd = (t < 7) ? d[t + 1] : rd;
      const float vm1 = v[t] + prv;
      const float grow = (vm1 > 0.0f) ? 1.0f : 0.0f;
      nd[t] = d[t] * v[t] +
              (d[t] + prd) * __builtin_amdgcn_rcpf(vm1 + EPS_DIV) * (grow - v[t]);
      nv[t] = grow;
      ch |= (grow != v[t]);
    }
#pragma unroll
    for (int t = 0; t < 8; ++t) { v[t] = nv[t]; d[t] = nd[t]; }
    if (__ballot(ch) == 0) break;
  }

  float4 o0, o1;
  float o[8];
#pragma unroll
  for (int t = 0; t < 8; ++t) o[t] = di[t] * v0[t] + d[t] * (1.0f - v0[t]);
  o0 = make_float4(o[0], o[1], o[2], o[3]);
  o1 = make_float4(o[4], o[5], o[6], o[7]);
  float* op = out + (size_t)row * WW + 8 * lane;
  *(float4*)(op)     = o0;
  *(float4*)(op + 4) = o1;
}

extern "C" void kernel_launch(void* const* d_in, const int* in_sizes, int n_in,
                              void* d_out, int out_size, void* d_ws, size_t ws_size,
                              hipStream_t stream) {
  const float* cr2l = (const float*)d_in[0];
  const float* cl2r = (const float*)d_in[1];
  const int*   md   = (const int*)d_in[2];

  const int BHW = out_size;      // B*H*W  (131072)
  const int BH  = BHW / WW;      // B*H    (512)

  float* vmbuf = (float*)d_ws;   // [BH, W]
  float* dini  = vmbuf + BHW;    // [BH, W]

  colsum_vm_kernel<<<BH, 256, 0, stream>>>(cl2r, md, vmbuf);
  dispini_kernel<<<BHW / 8, 256, 0, stream>>>(cr2l, md, dini);
  propagate_kernel<<<(BH + 7) / 8, 256, 0, stream>>>(dini, vmbuf, (float*)d_out, BH);
}